// FactorGraphMsgPassingLayer_NoDoubleCounting_71940702208298
// MI455X (gfx1250) — compile-verified
//
#include <hip/hip_runtime.h>
#include <hip/hip_bf16.h>

// Problem constants (match reference)
#define D_DIM 5
#define S_DIM 32
#define F_NUM 100000
#define V_NUM 60000
#define E_NUM (F_NUM * D_DIM)
#define TILES (F_NUM / 16)   // 6250, exact

typedef float v2f32 __attribute__((ext_vector_type(2)));
typedef float v4f32 __attribute__((ext_vector_type(4)));
typedef float v8f32 __attribute__((ext_vector_type(8)));

// ws float layout:
//   [0    .. 1024)  M  = W1^T W2^T
//   [1024 .. 2048)  M2 = W3^T W4^T
//   [2048 .. 2080)  c  = b1 W2^T + b2
//   [2080 .. 2112)  c2 = b3 W4^T + b4
//   [2112 .. 4160)  B-swizzled copy of M/M2: float4 per (k-chunk, matrix, lane)
//                   entry ((k*2+m)*32 + lane) = {Bt0.x, Bt0.y, Bt1.x, Bt1.y}
#define WS_BSWZ 2112

// ---------------------------------------------------------------------------
// Kernel P: fold the linear MLPs, then emit a WMMA-B-layout swizzled copy so
// k_fb can read its B operands as dense, coalesced b128 loads.
// ---------------------------------------------------------------------------
__global__ void __launch_bounds__(1024) k_prep(
        const float* __restrict__ W1, const float* __restrict__ b1,
        const float* __restrict__ W2, const float* __restrict__ b2,
        const float* __restrict__ W3, const float* __restrict__ b3,
        const float* __restrict__ W4, const float* __restrict__ b4,
        float* __restrict__ ws) {
    __shared__ float sm[2][1024];          // staged M, M2 for phase 2
    int t = threadIdx.x;                   // 1024 threads
    int i = t >> 5, j = t & 31;
    float m = 0.f, m2 = 0.f;
    #pragma unroll
    for (int k = 0; k < 32; ++k) {
        m  += W1[k * 32 + i] * W2[j * 32 + k];   // (W1^T W2^T)[i][j]
        m2 += W3[k * 32 + i] * W4[j * 32 + k];
    }
    ws[i * 32 + j]        = m;
    ws[1024 + i * 32 + j] = m2;
    sm[0][i * 32 + j] = m;
    sm[1][i * 32 + j] = m2;
    if (i == 0) {
        float c = 0.f, c2 = 0.f;
        #pragma unroll
        for (int k = 0; k < 32; ++k) {
            c  += b1[k] * W2[j * 32 + k];
            c2 += b3[k] * W4[j * 32 + k];
        }
        ws[2048 + j] = c + b2[j];
        ws[2080 + j] = c2 + b4[j];
    }
    __syncthreads();
    // phase 2: 512 threads each emit one float4 of the B-swizzled table
    if (t < 512) {
        const int k    = t >> 6;           // chunk 0..7
        const int mm   = (t >> 5) & 1;     // matrix select (M or M2)
        const int lane = t & 31;
        const int r  = lane & 15;
        const int h  = lane >> 4;
        const int kk = 4 * k + 2 * h;
        v4f32 q;
        q.x = sm[mm][kk * 32 + r];          // N-tile 0, K = kk
        q.y = sm[mm][(kk + 1) * 32 + r];    // N-tile 0, K = kk+1
        q.z = sm[mm][kk * 32 + 16 + r];     // N-tile 1, K = kk
        q.w = sm[mm][(kk + 1) * 32 + 16 + r];
        ((v4f32*)(ws + WS_BSWZ))[(k * 2 + mm) * 32 + lane] = q;
    }
}

__global__ void k_zero(float* __restrict__ p, int n) {
    int i = blockIdx.x * blockDim.x + threadIdx.x;
    if (i < n) p[i] = 0.f;
}

// ---------------------------------------------------------------------------
// Kernel A: factor->var messages + var_beliefs scatter-add.
// One wave32 per factor; lane = flattened state s. Group logsumexp over the
// 16 states sharing bit b at position p=4-d via 4 xor-shuffles (wave32).
// ---------------------------------------------------------------------------
__global__ void __launch_bounds__(256) k_f2v(
        const float* __restrict__ prv_v2f,   // [E,2]
        const float* __restrict__ prv_f2v,   // [E,2]
        const float* __restrict__ fb,        // [F,32]
        const int*   __restrict__ edge_v,    // [E]
        float* __restrict__ f2v_out,         // [E,2]
        float* __restrict__ var_beliefs)     // [V,2], pre-zeroed
{
    int f    = blockIdx.x * (blockDim.x >> 5) + (threadIdx.x >> 5);
    int lane = threadIdx.x & 31;
    if (f >= F_NUM) return;

    float fbv = fb[f * 32 + lane];           // coalesced 128B per wave

    #pragma unroll
    for (int d = 0; d < D_DIM; ++d) {
        const int e   = f * D_DIM + d;
        const int p   = 4 - d;               // bit position of this edge's var
        const int bit = (lane >> p) & 1;
        float m0 = prv_v2f[2 * e + 0];       // wave-uniform -> s_load
        float m1 = prv_v2f[2 * e + 1];
        float adj = fbv - (bit ? m1 : m0);

        // logsumexp over the 16-lane group sharing `bit` at position p
        float mx = adj;
        #pragma unroll
        for (int mb = 0; mb < 5; ++mb) {
            if (mb == p) continue;
            mx = fmaxf(mx, __shfl_xor(mx, 1 << mb, 32));
        }
        float t = __expf(adj - mx);
        #pragma unroll
        for (int mb = 0; mb < 5; ++mb) {
            if (mb == p) continue;
            t += __shfl_xor(t, 1 << mb, 32);
        }
        float lse = mx + __logf(t);

        float prior = bit ? prv_f2v[2 * e + 1] : prv_f2v[2 * e + 0];
        float msg = 0.5f * lse + 0.5f * prior;

        // normalize across the two components (partner lane differs in bit p)
        float other = __shfl_xor(msg, 1 << p, 32);
        float nm    = fmaxf(msg, other);
        msg -= nm + __logf(__expf(msg - nm) + __expf(other - nm));

        // one representative lane per (edge, bit): lane with all other bits 0
        if ((lane & ~(1 << p)) == 0) {
            f2v_out[2 * e + bit] = msg;
            atomicAdd(&var_beliefs[2 * edge_v[e] + bit], msg);
        }
    }
}

// ---------------------------------------------------------------------------
// Kernel B: var->factor messages, one thread per edge (b64 vectorized).
// ---------------------------------------------------------------------------
__global__ void k_v2f(const float* __restrict__ f2v,          // [E,2]
                      const float* __restrict__ var_beliefs,  // [V,2]
                      const int*   __restrict__ edge_v,
                      float* __restrict__ v2f_out)            // [E,2]
{
    int e = blockIdx.x * blockDim.x + threadIdx.x;
    if (e >= E_NUM) return;
    int v = edge_v[e];
    v2f32 vb = ((const v2f32*)var_beliefs)[v];
    v2f32 fv = ((const v2f32*)f2v)[e];
    float a = vb.x - fv.x;
    float b = vb.y - fv.y;
    float m = fmaxf(a, b);
    float norm = m + __logf(__expf(a - m) + __expf(b - m));
    v2f32 o; o.x = a - norm; o.y = b - norm;
    ((v2f32*)v2f_out)[e] = o;
}

// ---------------------------------------------------------------------------
// Kernel C: factor beliefs via folded MLPs on WMMA f32 path.
//   y[f,s]   = sum_d v2f[5f+d, bit_d(s)]        (built on the fly, A layout)
//   fb_pre   = pot + 0.5*y*M + 0.5*y + 2.5*c
//   fb_out   = 0.5*fb_pre*M2 + 0.5*fb_pre + 0.5*c2
// One wave per 16-factor tile; [16,32] output = 2 N-tiles x 8 chained
// v_wmma_f32_16x16x4_f32. LDS (16x33) bounces C layout -> A layout for GEMM2.
// B operands come pre-swizzled from ws: 16 coalesced b128 loads per lane.
// ---------------------------------------------------------------------------
__global__ void __launch_bounds__(256) k_fb(
        const float* __restrict__ v2f,   // [E,2]
        const float* __restrict__ pot,   // [F,32]
        const float* __restrict__ ws,    // M, M2, c, c2, B-swizzled table
        float* __restrict__ fb_out)      // [F,32]
{
    __shared__ float lds[8][16][33];     // +1 pad vs 64 banks
    const int wave = threadIdx.x >> 5;
    const int lane = threadIdx.x & 31;
    int tile = blockIdx.x * 8 + wave;
    if (tile >= TILES) tile = TILES - 1; // duplicate (identical) work, keeps barriers uniform
    const int f0 = tile * 16;
    const int r  = lane & 15;            // A-layout row / B-layout N index
    const int h  = lane >> 4;            // half-wave select

    // --- per-lane messages for factor f0+r: 5 contiguous float2 (b64) ---
    const v2f32* pm = (const v2f32*)v2f;
    v2f32 msg[D_DIM];
    const int fr = f0 + r;
    #pragma unroll
    for (int d = 0; d < D_DIM; ++d)
        msg[d] = pm[D_DIM * fr + d];

    // --- build Y in WMMA A layout (16x4 f32 chunks); mirror rows into LDS ---
    v2f32 A1[8];
    #pragma unroll
    for (int k = 0; k < 8; ++k) {
        const int s0 = 4 * k + 2 * h;
        float y0 = 0.f, y1 = 0.f;
        #pragma unroll
        for (int d = 0; d < D_DIM; ++d) {
            y0 += ((s0 >> (4 - d)) & 1) ? msg[d].y : msg[d].x;
            y1 += (((s0 + 1) >> (4 - d)) & 1) ? msg[d].y : msg[d].x;
        }
        A1[k].x = y0; A1[k].y = y1;
        lds[wave][r][s0]     = y0;
        lds[wave][r][s0 + 1] = y1;
    }

    // --- B operands: pre-swizzled, 16 dense b128 loads per lane ---
    const v4f32* bsw = (const v4f32*)(ws + WS_BSWZ);
    v2f32 B1[2][8], B2[2][8];
    #pragma unroll
    for (int k = 0; k < 8; ++k) {
        v4f32 q1 = bsw[(k * 2 + 0) * 32 + lane];
        v4f32 q2 = bsw[(k * 2 + 1) * 32 + lane];
        B1[0][k].x = q1.x; B1[0][k].y = q1.y;
        B1[1][k].x = q1.z; B1[1][k].y = q1.w;
        B2[0][k].x = q2.x; B2[0][k].y = q2.y;
        B2[1][k].x = q2.z; B2[1][k].y = q2.w;
    }
    const float cvec[2]  = { ws[2048 + r], ws[2048 + 16 + r] };
    const float c2vec[2] = { ws[2080 + r], ws[2080 + 16 + r] };

    // --- GEMM1: Y @ M, K=32 via 8 chained 16x16x4 f32 WMMAs per N-tile ---
    v8f32 acc0 = {}, acc1 = {};
    #pragma unroll
    for (int k = 0; k < 8; ++k) {
        acc0 = __builtin_amdgcn_wmma_f32_16x16x4_f32(false, A1[k], false, B1[0][k],
                                                     (short)0, acc0, false, false);
        acc1 = __builtin_amdgcn_wmma_f32_16x16x4_f32(false, A1[k], false, B1[1][k],
                                                     (short)0, acc1, false, false);
    }

    __syncthreads();   // Y rows fully resident in LDS

    // --- fb_pre in C layout: row = j + 8h, col = 16t + r ---
    float fbpre[2][8];
    #pragma unroll
    for (int j = 0; j < 8; ++j) {
        const int row = j + 8 * h;
        const float a[2] = { acc0[j], acc1[j] };
        #pragma unroll
        for (int t = 0; t < 2; ++t) {
            const int col = 16 * t + r;
            const float y = lds[wave][row][col];
            const float p = pot[(f0 + row) * 32 + col];
            fbpre[t][j] = p + 0.5f * a[t] + 0.5f * y + 2.5f * cvec[t];
        }
    }

    __syncthreads();   // done reading Y; reuse LDS for transpose
    #pragma unroll
    for (int j = 0; j < 8; ++j) {
        const int row = j + 8 * h;
        lds[wave][row][r]      = fbpre[0][j];
        lds[wave][row][16 + r] = fbpre[1][j];
    }
    __syncthreads();

    // --- GEMM2: fb_pre @ M2, A read back in A layout ---
    v8f32 acd0 = {}, acd1 = {};
    #pragma unroll
    for (int k = 0; k < 8; ++k) {
        v2f32 a;
        a.x = lds[wave][r][4 * k + 2 * h];
        a.y = lds[wave][r][4 * k + 2 * h + 1];
        acd0 = __builtin_amdgcn_wmma_f32_16x16x4_f32(false, a, false, B2[0][k],
                                                     (short)0, acd0, false, false);
        acd1 = __builtin_amdgcn_wmma_f32_16x16x4_f32(false, a, false, B2[1][k],
                                                     (short)0, acd1, false, false);
    }

    // --- residual + write out (C layout: 64B-contiguous lane groups) ---
    #pragma unroll
    for (int j = 0; j < 8; ++j) {
        const int row = j + 8 * h;
        const float a[2] = { acd0[j], acd1[j] };
        #pragma unroll
        for (int t = 0; t < 2; ++t) {
            const int col = 16 * t + r;
            fb_out[(f0 + row) * 32 + col] = 0.5f * a[t] + 0.5f * fbpre[t][j] + 0.5f * c2vec[t];
        }
    }
}

// ---------------------------------------------------------------------------
extern "C" void kernel_launch(void* const* d_in, const int* in_sizes, int n_in,
                              void* d_out, int out_size, void* d_ws, size_t ws_size,
                              hipStream_t stream) {
    const float* prv_v2f = (const float*)d_in[0];
    const float* prv_f2v = (const float*)d_in[1];
    const float* prv_fb  = (const float*)d_in[2];
    const float* pot     = (const float*)d_in[3];
    const float* W1 = (const float*)d_in[4];  const float* b1 = (const float*)d_in[5];
    const float* W2 = (const float*)d_in[6];  const float* b2 = (const float*)d_in[7];
    const float* W3 = (const float*)d_in[8];  const float* b3 = (const float*)d_in[9];
    const float* W4 = (const float*)d_in[10]; const float* b4 = (const float*)d_in[11];
    const int* edge_v = (const int*)d_in[13];
    (void)in_sizes; (void)n_in; (void)out_size; (void)ws_size;

    float* out = (float*)d_out;
    // output layout: v2f[E,2] | f2v[E,2] | factor_beliefs[F,32] | var_beliefs[V,2]
    float* o_v2f = out;
    float* o_f2v = out + (size_t)E_NUM * 2;
    float* o_fb  = out + (size_t)E_NUM * 4;
    float* o_vb  = out + (size_t)E_NUM * 4 + (size_t)F_NUM * S_DIM;
    float* ws    = (float*)d_ws;  // needs 4160 floats (16.3 KB)

    k_prep<<<1, 1024, 0, stream>>>(W1, b1, W2, b2, W3, b3, W4, b4, ws);
    k_zero<<<(V_NUM * 2 + 255) / 256, 256, 0, stream>>>(o_vb, V_NUM * 2);
    k_f2v<<<F_NUM / 8, 256, 0, stream>>>(prv_v2f, prv_f2v, prv_fb, edge_v, o_f2v, o_vb);
    k_v2f<<<(E_NUM + 255) / 256, 256, 0, stream>>>(o_f2v, o_vb, edge_v, o_v2f);
    k_fb<<<(TILES + 7) / 8, 256, 0, stream>>>(o_v2f, pot, ws, o_fb);
}